// SpERT_67353677136076
// MI455X (gfx1250) — compile-verified
//
#include <hip/hip_runtime.h>
#include <hip/hip_bf16.h>
#include <stdint.h>

// ---- problem constants (from reference) ----
#define BB 4
#define LL 512
#define HH 768
#define SS 64
#define RR 128
#define EE 25
#define NEGV -1e20f

#define KE 800     // padded entity-repr width: 768 + 25 -> 800 (mult of 32)
#define KR 2368    // padded relation-repr width: 2354 -> 2368 (mult of 32)
#define NTILE 16   // WMMA N
#define KCHUNK 32  // K rows staged per async chunk (32*16 fp32 = 2KB)

typedef __attribute__((ext_vector_type(2))) float v2f;
typedef __attribute__((ext_vector_type(8))) float v8f;

// ------------------------------------------------------------------
// K1: entity masked max-pool + size embedding -> padded repr rows
// grid = B*S blocks, 256 threads
// ------------------------------------------------------------------
__global__ void k_entity_pool(const float* __restrict__ hidden,
                              const int* __restrict__ emasks,
                              const float* __restrict__ size_emb,
                              float* __restrict__ ereprs) {
    const int bs = blockIdx.x;           // b*S + s
    const int b  = bs / SS;
    const int t  = threadIdx.x;          // 0..255
    const int* mrow = emasks + bs * LL;
    const float* hbase = hidden + (long)b * LL * HH;

    float a0 = NEGV, a1 = NEGV, a2 = NEGV;
    int cnt = 0;
    for (int l = 0; l < LL; ++l) {
        int m = mrow[l];
        cnt += m;
        if (m) {
            const float* hp = hbase + l * HH;
            a0 = fmaxf(a0, hp[t]);
            a1 = fmaxf(a1, hp[t + 256]);
            a2 = fmaxf(a2, hp[t + 512]);
        }
    }
    if (cnt > 99) cnt = 99;
    if (cnt < 0)  cnt = 0;
    float* out = ereprs + (long)bs * KE;
    out[t]       = a0;
    out[t + 256] = a1;
    out[t + 512] = a2;
    if (t < EE)       out[HH + t] = size_emb[cnt * EE + t];
    else if (t < 32)  out[HH + t] = 0.0f;   // pad cols 793..799
}

// ------------------------------------------------------------------
// K2: relation context masked max-pool (+sample-mask zero fill)
// writes cols [0,768) of relation rows, zeroes pad cols [2354,2368)
// grid = B*R blocks, 256 threads
// ------------------------------------------------------------------
__global__ void k_ctx_pool(const float* __restrict__ hidden,
                           const int* __restrict__ cmasks,
                           const int* __restrict__ smask,
                           float* __restrict__ rreprs) {
    const int br = blockIdx.x;           // b*R + r
    const int b  = br / RR;
    const int t  = threadIdx.x;
    const int* mrow = cmasks + br * LL;
    const float* hbase = hidden + (long)b * LL * HH;

    float a0 = NEGV, a1 = NEGV, a2 = NEGV;
    for (int l = 0; l < LL; ++l) {
        if (mrow[l]) {
            const float* hp = hbase + l * HH;
            a0 = fmaxf(a0, hp[t]);
            a1 = fmaxf(a1, hp[t + 256]);
            a2 = fmaxf(a2, hp[t + 512]);
        }
    }
    if (smask[br] == 1) { a0 = 0.0f; a1 = 0.0f; a2 = 0.0f; }
    float* row = rreprs + (long)br * KR;
    row[t]       = a0;
    row[t + 256] = a1;
    row[t + 512] = a2;
    if (t < (KR - 2354)) row[2354 + t] = 0.0f;   // pad cols
}

// ------------------------------------------------------------------
// K3: gather head/tail entity reprs + sizes into relation rows
// grid = B*R blocks, 256 threads
// ------------------------------------------------------------------
__global__ void k_pair_gather(const int* __restrict__ relations,
                              const float* __restrict__ ereprs,
                              float* __restrict__ rreprs) {
    const int br = blockIdx.x;
    const int b  = br / RR;
    const int t  = threadIdx.x;
    int head = relations[br * 2 + 0];
    int tail = relations[br * 2 + 1];
    if (head < 0) head = 0; if (head >= SS) head = SS - 1;
    if (tail < 0) tail = 0; if (tail >= SS) tail = SS - 1;
    const float* eh = ereprs + (long)(b * SS + head) * KE;
    const float* et = ereprs + (long)(b * SS + tail) * KE;
    float* row = rreprs + (long)br * KR;
    for (int i = t; i < HH; i += 256) {
        row[768 + i]  = eh[i];
        row[1536 + i] = et[i];
    }
    if (t < EE) {
        row[2304 + t] = eh[HH + t];
        row[2329 + t] = et[HH + t];
    }
}

// ------------------------------------------------------------------
// K4: build padded, transposed weight matrices  B[k][n] = W[n][k]
// ------------------------------------------------------------------
__global__ void k_pack_weights(const float* __restrict__ span_W,
                               const float* __restrict__ rel_W,
                               float* __restrict__ Bspan,
                               float* __restrict__ Brel) {
    const int NS = KE * NTILE;           // 12800
    const int NR = KR * NTILE;           // 37888
    for (int i = blockIdx.x * blockDim.x + threadIdx.x; i < NS + NR;
         i += gridDim.x * blockDim.x) {
        if (i < NS) {
            int k = i / NTILE, n = i % NTILE;
            Bspan[i] = (n < 8 && k < 793) ? span_W[n * 793 + k] : 0.0f;
        } else {
            int j = i - NS;
            int k = j / NTILE, n = j % NTILE;
            Brel[j] = (n < 5 && k < 2354) ? rel_W[n * 2354 + k] : 0.0f;
        }
    }
}

// ------------------------------------------------------------------
// K5/K6: fp32 WMMA GEMM  out[M x Nout] = A[M x K] * Bm[K x 16] + bias
// one wave (32 threads) per 16-row M tile; K multiple of 32.
// B panel is double-buffered through LDS with global_load_async_to_lds_b128,
// synchronized with s_wait_asynccnt (async loads complete in order).
// A frag (16x4 f32): lanes 0-15 M=lane, K={k,k+1}; lanes 16-31 K={k+2,k+3}
// B frag (4x16 f32): lanes 0-15 N=lane, K rows {k,k+1}; lanes 16-31 {k+2,k+3}
// D (16x16 f32): vgpr r -> M = r + 8*(lane>=16), N = lane&15
// ------------------------------------------------------------------
__global__ void k_wmma_gemm(const float* __restrict__ A,
                            const float* __restrict__ Bm,
                            const float* __restrict__ bias,
                            float* __restrict__ out,
                            int K, int Nout) {
    __shared__ float lb[2][KCHUNK * NTILE];   // 2 x 2KB double buffer

    const int tile = blockIdx.x;
    const int lane = threadIdx.x & 31;
    const int hi   = lane >> 4;          // 0: K pair {0,1}, 1: {2,3}
    const int mr   = lane & 15;          // A row within tile / B column
    const float* Arow = A + (long)(tile * 16 + mr) * K;

    // async-stage one 32x16 fp32 B chunk into LDS buffer `buf`
    auto stage = [&](int chunk, int buf) {
        const float* src = Bm + (long)chunk * (KCHUNK * NTILE);
#pragma unroll
        for (int j = 0; j < 4; ++j) {
            unsigned lds_addr =
                (unsigned)(uintptr_t)(&lb[buf][(j * 32 + lane) * 4]);
            const float* g = src + (j * 32 + lane) * 4;
            asm volatile("global_load_async_to_lds_b128 %0, %1, off"
                         :: "v"(lds_addr), "v"(g) : "memory");
        }
    };

    const int nchunk = K / KCHUNK;
    stage(0, 0);

    v8f acc = {};
    for (int c = 0; c < nchunk; ++c) {
        const int buf = c & 1;
        if (c + 1 < nchunk) {
            stage(c + 1, buf ^ 1);
            // chunk c's 4 issues complete before chunk c+1's 4 (in-order)
            asm volatile("s_wait_asynccnt 0x4" ::: "memory");
        } else {
            asm volatile("s_wait_asynccnt 0x0" ::: "memory");
        }
        const float* Ab = Arow + c * KCHUNK;
        __builtin_prefetch((const void*)(Ab + 2 * KCHUNK), 0, 1);
#pragma unroll
        for (int k2 = 0; k2 < KCHUNK; k2 += 4) {
            const int kk = k2 + 2 * hi;
            v2f a, bf;
            a.x  = Ab[kk];
            a.y  = Ab[kk + 1];
            bf.x = lb[buf][kk * NTILE + mr];
            bf.y = lb[buf][(kk + 1) * NTILE + mr];
            acc = __builtin_amdgcn_wmma_f32_16x16x4_f32(
                /*neg_a=*/false, a, /*neg_b=*/false, bf,
                /*c_mod=*/(short)0, acc, /*reuse_a=*/false, /*reuse_b=*/false);
        }
    }

    const int n = mr;
    if (n < Nout) {
        const float bv = bias[n];
#pragma unroll
        for (int r = 0; r < 8; ++r) {
            const int m = tile * 16 + r + hi * 8;
            out[m * Nout + n] = acc[r] + bv;
        }
    }
}

// ------------------------------------------------------------------
extern "C" void kernel_launch(void* const* d_in, const int* in_sizes, int n_in,
                              void* d_out, int out_size, void* d_ws, size_t ws_size,
                              hipStream_t stream) {
    const float* hidden   = (const float*)d_in[0];
    const int*   emasks   = (const int*)  d_in[1];
    const int*   relations= (const int*)  d_in[2];
    const int*   cmasks   = (const int*)  d_in[3];
    const int*   smask    = (const int*)  d_in[4];
    const float* size_emb = (const float*)d_in[5];
    const float* span_W   = (const float*)d_in[6];
    const float* span_b   = (const float*)d_in[7];
    const float* rel_W    = (const float*)d_in[8];
    const float* rel_b    = (const float*)d_in[9];

    float* out = (float*)d_out;
    float* ent_logits = out;                // [4,64,8]  = 2048
    float* rel_logits = out + BB * SS * 8;  // [4,128,5] = 2560

    // workspace carve-up (floats)
    float* ws = (float*)d_ws;
    float* ereprs = ws;                                   // 256*800
    float* rreprs = ereprs + (size_t)BB * SS * KE;        // 512*2368
    float* Bspan  = rreprs + (size_t)BB * RR * KR;        // 800*16
    float* Brel   = Bspan + (size_t)KE * NTILE;           // 2368*16

    k_entity_pool<<<BB * SS, 256, 0, stream>>>(hidden, emasks, size_emb, ereprs);
    k_ctx_pool<<<BB * RR, 256, 0, stream>>>(hidden, cmasks, smask, rreprs);
    k_pair_gather<<<BB * RR, 256, 0, stream>>>(relations, ereprs, rreprs);
    k_pack_weights<<<198, 256, 0, stream>>>(span_W, rel_W, Bspan, Brel);

    // entity GEMM: M=256 -> 16 tiles, K=800, Nout=8
    k_wmma_gemm<<<(BB * SS) / 16, 32, 0, stream>>>(ereprs, Bspan, span_b,
                                                   ent_logits, KE, 8);
    // relation GEMM: M=512 -> 32 tiles, K=2368, Nout=5
    k_wmma_gemm<<<(BB * RR) / 16, 32, 0, stream>>>(rreprs, Brel, rel_b,
                                                   rel_logits, KR, 5);
}